// GaussianAdapter_38199439131199
// MI455X (gfx1250) — compile-verified
//
#include <hip/hip_runtime.h>
#include <cstdint>

// Problem constants from the reference: (B, T, R, F) = (16, 512, 128, 32)
constexpr int B = 16;
constexpr int T = 512;
constexpr int R = 128;
constexpr int F = 32;

constexpr int CH  = 64;                  // timesteps staged per chunk
constexpr int NC  = T / CH;              // 8 chunks
constexpr int THREADS = 256;             // 8 wave32 per block
constexpr int SEGS = (CH * F * 4) / 16;  // 512 x b128 segments per array per chunk
constexpr int SEG_PER_THREAD = SEGS / THREADS;  // 2

#define EPS_F 1e-7f

// CDNA5 async DMA: global -> LDS, 16 bytes per lane, tracked by ASYNCcnt.
__device__ __forceinline__ void async_b128_to_lds(uint32_t lds_byte_addr, const void* gaddr) {
    asm volatile("global_load_async_to_lds_b128 %0, %1, off"
                 :: "v"(lds_byte_addr), "v"(gaddr)
                 : "memory");
}

__device__ __forceinline__ void wait_asynccnt0() {
    asm volatile("s_wait_asynccnt 0x0" ::: "memory");
}

__global__ __launch_bounds__(THREADS)
void GaussianAdapter_kernel(const float* __restrict__ tsteps,
                            const float* __restrict__ values,
                            const float* __restrict__ refts,
                            const float* __restrict__ alpha,
                            float* __restrict__ out) {
    // Double-buffered staging tiles: 2 bufs x 2 arrays x 8KB = 32KB LDS.
    __shared__ __align__(16) float s_ts[2][CH * F];
    __shared__ __align__(16) float s_vl[2][CH * F];

    const int tid  = threadIdx.x;
    const int lane = tid & 31;          // lane == feature index f
    const int wave = tid >> 5;          // 8 waves -> 8 consecutive r
    const int b    = blockIdx.x >> 4;   // 16 blocks per batch
    const int r    = ((blockIdx.x & 15) << 3) + wave;

    const float a   = alpha[0];
    const float c2  = -a * 1.44269504088896340736f;  // -alpha * log2(e)
    const float rt  = refts[b * R + r];

    const float* gts = tsteps + (size_t)b * T * F;
    const float* gvl = values + (size_t)b * T * F;

    // LDS byte addresses (low 32 bits of the generic shared pointer).
    uint32_t ts_off[2], vl_off[2];
    ts_off[0] = (uint32_t)(uintptr_t)&s_ts[0][0];
    ts_off[1] = (uint32_t)(uintptr_t)&s_ts[1][0];
    vl_off[0] = (uint32_t)(uintptr_t)&s_vl[0][0];
    vl_off[1] = (uint32_t)(uintptr_t)&s_vl[1][0];

    // Cooperative async issue of one chunk into buffer p (4 b128 ops per thread).
    auto issue_chunk = [&](int c, int p) {
        const float* gts_c = gts + c * (CH * F);
        const float* gvl_c = gvl + c * (CH * F);
#pragma unroll
        for (int s = 0; s < SEG_PER_THREAD; ++s) {
            const int seg = tid + s * THREADS;            // 16-byte segment index
            async_b128_to_lds(ts_off[p] + seg * 16, gts_c + seg * 4);
            async_b128_to_lds(vl_off[p] + seg * 16, gvl_c + seg * 4);
        }
    };

    issue_chunk(0, 0);

    float num = 0.0f;
    float den = 0.0f;

    for (int c = 0; c < NC; ++c) {
        const int p = c & 1;
        // Chunk c is the only outstanding async work for this wave.
        wait_asynccnt0();
        __syncthreads();                       // all waves' DMA landed in LDS
        if (c + 1 < NC) issue_chunk(c + 1, p ^ 1);  // overlap next DMA with compute

        const float* lts = s_ts[p];
        const float* lvl = s_vl[p];
#pragma unroll 4
        for (int t = 0; t < CH; ++t) {
            const float x = lts[t * F + lane];
            const float v = lvl[t * F + lane];
            const float d = rt - x;
            float e = __builtin_amdgcn_exp2f(c2 * d * d);  // v_exp_f32
            e = (x > 0.0f) ? e : 0.0f;                     // mask
            e += EPS_F;                                    // + EPS (the /R cancels)
            den += e;
            num = fmaf(e, v, num);
        }
        __syncthreads();   // buffer p may be refilled next iteration
    }

    out[((size_t)b * R + r) * F + lane] = num / den;
}

extern "C" void kernel_launch(void* const* d_in, const int* in_sizes, int n_in,
                              void* d_out, int out_size, void* d_ws, size_t ws_size,
                              hipStream_t stream) {
    const float* tsteps = (const float*)d_in[0];   // (B, T, F) f32
    const float* values = (const float*)d_in[1];   // (B, T, F) f32
    const float* refts  = (const float*)d_in[2];   // (B, R)    f32
    const float* alpha  = (const float*)d_in[3];   // (1,)      f32
    float* out = (float*)d_out;                    // (B, R, F) f32

    const dim3 grid(B * (R / 8));  // 256 blocks, 8 waves each -> one wave per (b, r)
    GaussianAdapter_kernel<<<grid, THREADS, 0, stream>>>(tsteps, values, refts, alpha, out);
}